// DNM_Linear_6176162971841
// MI455X (gfx1250) — compile-verified
//
#include <hip/hip_runtime.h>

typedef __attribute__((ext_vector_type(2))) float v2f;
typedef __attribute__((ext_vector_type(8))) float v8f;

namespace {
constexpr int kB   = 128;
constexpr int kIn  = 512;
constexpr int kOut = 256;
constexpr int kM   = 16;
// LDS row stride of 516 floats: lane b loads float2 at (b*516 + ii)*4B.
// 516 % 64 == 4 banks -> 16 lanes hit banks {4b, 4b+1}, all distinct. Conflict-free.
constexpr int kXsStride = kIn + 4;
}

// Each workgroup: 16 batches x 16 outputs. 8 waves; each wave owns 2 outputs
// and the full (m,i) reduction for its 16-batch tile.
// Per wmma: A[b, k] = relu(x[b, i0+k]*W[o,m,i0+k] - q[o,m,i0+k]), k = 0..3,
// B = ones(4x16)  =>  C[b, :] += sum_k A[b,k]   (f32-precise reduction on the
// matrix pipe, co-executing with the VALU fma/max stream).
__global__ __launch_bounds__(256, 1)
void dnm_linear_wmma(const float* __restrict__ x,
                     const float* __restrict__ W,
                     const float* __restrict__ q,
                     float* __restrict__ out)
{
    __shared__ float xs[16 * kXsStride];

    const int tid  = threadIdx.x;
    const int lane = tid & 31;
    const int wid  = tid >> 5;

    const int b_base = blockIdx.y * 16;          // batch tile
    const int o0     = blockIdx.x * 16 + wid * 2; // this wave's first output

    // Cooperative load of the x tile [16][512] into padded LDS (exactly 32 per thread).
    for (int idx = tid; idx < 16 * kIn; idx += 256) {
        const int b = idx >> 9;          // /512
        const int i = idx & (kIn - 1);   // %512
        xs[b * kXsStride + i] = x[(b_base + b) * kIn + i];
    }
    __syncthreads();

    // WMMA A-matrix lane mapping (32-bit A 16x4):
    //   lanes 0-15 : rows M=0..15, K=0,1   lanes 16-31: rows M=0..15, K=2,3
    const int b_lane = lane & 15;
    const int khalf  = lane >> 4;

    const float* w0 = W + (size_t)(o0    ) * kM * kIn;
    const float* w1 = W + (size_t)(o0 + 1) * kM * kIn;
    const float* q0 = q + (size_t)(o0    ) * kM * kIn;
    const float* q1 = q + (size_t)(o0 + 1) * kM * kIn;

    v8f c0 = {};   // f32 accumulator, rows = batches (columns redundant)
    v8f c1 = {};
    v2f ones;
    ones.x = 1.0f;
    ones.y = 1.0f;

    for (int i0 = 0; i0 < kIn; i0 += 4) {
        const int ii = i0 + khalf * 2;   // this lane's K-pair within the i0 block
        const float2 xv = *(const float2*)&xs[b_lane * kXsStride + ii];

        // token lookahead prefetch of next W/q lines (global_prefetch_b8)
        __builtin_prefetch(&w0[(kM - 1) * kIn + ii + 16], 0, 1);
        __builtin_prefetch(&q0[(kM - 1) * kIn + ii + 16], 0, 1);

        #pragma unroll
        for (int m = 0; m < kM; ++m) {
            const int off = m * kIn + ii;
            {
                const float2 wv = *(const float2*)&w0[off];
                const float2 qv = *(const float2*)&q0[off];
                v2f a;
                a.x = fmaxf(fmaf(xv.x, wv.x, -qv.x), 0.0f);
                a.y = fmaxf(fmaf(xv.y, wv.y, -qv.y), 0.0f);
                c0 = __builtin_amdgcn_wmma_f32_16x16x4_f32(false, a, false, ones,
                                                           (short)0, c0, false, false);
            }
            {
                const float2 wv = *(const float2*)&w1[off];
                const float2 qv = *(const float2*)&q1[off];
                v2f a;
                a.x = fmaxf(fmaf(xv.x, wv.x, -qv.x), 0.0f);
                a.y = fmaxf(fmaf(xv.y, wv.y, -qv.y), 0.0f);
                c1 = __builtin_amdgcn_wmma_f32_16x16x4_f32(false, a, false, ones,
                                                           (short)0, c1, false, false);
            }
        }
    }

    // C/D layout: VGPR r -> lanes 0-15 hold row M=r, lanes 16-31 hold row M=r+8.
    // All columns equal (B was ones), so lane 0 emits rows 0..7, lane 16 rows 8..15.
    // out = relu(K*(K*acc - QS)) = relu(0.25*acc - 0.05)
    if (b_lane == 0) {
        #pragma unroll
        for (int r = 0; r < 8; ++r) {
            const int b = b_base + khalf * 8 + r;
            out[b * kOut + o0]     = fmaxf(0.25f * c0[r] - 0.05f, 0.0f);
            out[b * kOut + o0 + 1] = fmaxf(0.25f * c1[r] - 0.05f, 0.0f);
        }
    }
}

extern "C" void kernel_launch(void* const* d_in, const int* in_sizes, int n_in,
                              void* d_out, int out_size, void* d_ws, size_t ws_size,
                              hipStream_t stream) {
    (void)in_sizes; (void)n_in; (void)d_ws; (void)ws_size; (void)out_size;
    const float* x = (const float*)d_in[0];
    const float* W = (const float*)d_in[1];
    const float* q = (const float*)d_in[2];
    float* out     = (float*)d_out;

    dim3 grid(kOut / 16, kB / 16);   // 16 x 8 = 128 workgroups
    dim3 block(256);                 // 8 waves (wave32)
    hipLaunchKernelGGL(dnm_linear_wmma, grid, block, 0, stream, x, W, q, out);
}